// GCN_31430570672165
// MI455X (gfx1250) — compile-verified
//
#include <hip/hip_runtime.h>

typedef float v2f __attribute__((ext_vector_type(2)));
typedef float v8f __attribute__((ext_vector_type(8)));

// ---------------------------------------------------------------------------
// Dense GEMM H = relu?(X) @ W using fp32 WMMA (V_WMMA_F32_16X16X4_F32).
// X: M x K row-major, W: K x N row-major, H: M x N row-major.
// K, N, RELU are compile-time -> immediate-offset loads, no 64-bit addr mul.
// One wave computes a 16(M) x 64(N) strip: 4 accumulators reuse one A frag.
// All control flow is wave-uniform -> EXEC all ones for WMMA (ISA req).
// ---------------------------------------------------------------------------
template <int K, int N, bool RELU>
__global__ void gcn_gemm_f32(const float* __restrict__ X,
                             const float* __restrict__ W,
                             float* __restrict__ H,
                             int M)
{
    const int lane  = threadIdx.x & 31;
    const int wave  = (blockIdx.x * blockDim.x + threadIdx.x) >> 5;
    const int ngrp  = N >> 6;                 // 64-wide N groups per M-tile row
    const int mtile = wave / ngrp;
    const int ng    = wave % ngrp;
    if (mtile * 16 >= M) return;              // wave-uniform guard

    const int row   = lane & 15;              // M row within tile (A), N col (B/C)
    const int hi    = lane >> 4;              // half-wave select
    const int arow  = mtile * 16 + row;
    const int ncol0 = ng * 64;

    v8f acc0 = {}, acc1 = {}, acc2 = {}, acc3 = {};

    // A VGPR0/1 hold K = (k0 + 2*hi) and +1  (ISA 32-bit A 16x4 layout)
    const float* Xp = X + (size_t)arow * K + hi * 2;
    // B VGPR0/1 hold rows K = (k0 + 2*hi) and +1 at column (ncol0 + j*16 + row)
    const float* Wp = W + (size_t)(hi * 2) * N + ncol0 + row;

#pragma unroll 4
    for (int k0 = 0; k0 < K; k0 += 4) {
        v2f a = *(const v2f*)Xp;
        Xp += 4;
        if (RELU) { a.x = fmaxf(a.x, 0.0f); a.y = fmaxf(a.y, 0.0f); }

        v2f b0 = { Wp[0],  Wp[N + 0]  };
        v2f b1 = { Wp[16], Wp[N + 16] };
        v2f b2 = { Wp[32], Wp[N + 32] };
        v2f b3 = { Wp[48], Wp[N + 48] };
        Wp += 4 * N;

        acc0 = __builtin_amdgcn_wmma_f32_16x16x4_f32(false, a, false, b0, (short)0, acc0, false, false);
        acc1 = __builtin_amdgcn_wmma_f32_16x16x4_f32(false, a, false, b1, (short)0, acc1, false, false);
        acc2 = __builtin_amdgcn_wmma_f32_16x16x4_f32(false, a, false, b2, (short)0, acc2, false, false);
        acc3 = __builtin_amdgcn_wmma_f32_16x16x4_f32(false, a, false, b3, (short)0, acc3, false, false);
    }

    // C/D layout: VGPR r -> row (r + 8*hi), col = lane&15
    float* Hp = H + (size_t)(mtile * 16 + hi * 8) * N + ncol0 + row;
#pragma unroll
    for (int r = 0; r < 8; ++r) {
        Hp[r * N + 0 ] = acc0[r];
        Hp[r * N + 16] = acc1[r];
        Hp[r * N + 32] = acc2[r];
        Hp[r * N + 48] = acc3[r];
    }
}

// ---------------------------------------------------------------------------
// Degree / normalization helpers
// ---------------------------------------------------------------------------
__global__ void gcn_deg_init(float* __restrict__ deg, int n)
{
    int i = blockIdx.x * blockDim.x + threadIdx.x;
    if (i < n) deg[i] = 1.0f;                 // self-loop contributes 1 per node
}

__global__ void gcn_deg_count(const int* __restrict__ dst, float* __restrict__ deg, int E)
{
    int e = blockIdx.x * blockDim.x + threadIdx.x;
    if (e < E) unsafeAtomicAdd(&deg[dst[e]], 1.0f);
}

__global__ void gcn_dinv(const float* __restrict__ deg, float* __restrict__ dinv, int n)
{
    int i = blockIdx.x * blockDim.x + threadIdx.x;
    if (i < n) {
        float d = deg[i];
        dinv[i] = (d > 0.0f) ? rsqrtf(d) : 0.0f;
    }
}

// out[i,f] = bias[f] + H[i,f] * dinv[i]^2   (bias + fused self-loop message)
template <int F>
__global__ void gcn_init_out(const float* __restrict__ H,
                             const float* __restrict__ bias,
                             const float* __restrict__ dinv,
                             float* __restrict__ out,
                             int n)
{
    long idx = (long)blockIdx.x * blockDim.x + threadIdx.x;
    if (idx >= (long)n * F) return;
    int i = (int)(idx / F);       // F is a power of two -> shift/mask
    int f = (int)(idx & (F - 1));
    float di = dinv[i];
    out[idx] = bias[f] + H[idx] * (di * di);
}

// ---------------------------------------------------------------------------
// Edge scatter: one wave per edge; lanes stream float4 chunks of the feature
// row; fp32 hardware atomics (global_atomic_add_f32) accumulate into out[dst].
// src/dst are wave-uniform: readfirstlane puts row bases in SGPRs so the
// gather/atomic address math is scalar (saddr + lane offset form).
// Both H and out fit in the 192MB L2, so this runs at L2 atomic bandwidth.
// ---------------------------------------------------------------------------
template <int F>
__global__ void gcn_scatter(const float* __restrict__ H,
                            const int* __restrict__ src,
                            const int* __restrict__ dst,
                            const float* __restrict__ dinv,
                            float* __restrict__ out, int E)
{
    int gid  = blockIdx.x * blockDim.x + threadIdx.x;
    int e    = gid >> 5;
    int lane = gid & 31;
    if (e >= E) return;

    int s = __builtin_amdgcn_readfirstlane(src[e]);   // wave-uniform -> SGPR
    int d = __builtin_amdgcn_readfirstlane(dst[e]);
    float nrm = dinv[s] * dinv[d];

    const float4* hp = (const float4*)(H + (size_t)s * F);
    float* op = out + (size_t)d * F;

#pragma unroll
    for (int i = 0; i < F / 128; ++i) {
        float4 v = hp[lane + i * 32];
        int f = (lane + i * 32) * 4;
        unsafeAtomicAdd(op + f + 0, v.x * nrm);
        unsafeAtomicAdd(op + f + 1, v.y * nrm);
        unsafeAtomicAdd(op + f + 2, v.z * nrm);
        unsafeAtomicAdd(op + f + 3, v.w * nrm);
    }
}

// ---------------------------------------------------------------------------
// Host-side orchestration
// ---------------------------------------------------------------------------
extern "C" void kernel_launch(void* const* d_in, const int* in_sizes, int n_in,
                              void* d_out, int out_size, void* d_ws, size_t ws_size,
                              hipStream_t stream)
{
    const int D_IN = 256, D_H = 256, D_OUT = 128;

    const float* x   = (const float*)d_in[0];
    const int*   ei  = (const int*)  d_in[1];
    const float* W1  = (const float*)d_in[2];
    const float* b1  = (const float*)d_in[3];
    const float* W2  = (const float*)d_in[4];
    const float* b2  = (const float*)d_in[5];
    const float* W3  = (const float*)d_in[6];
    const float* b3  = (const float*)d_in[7];
    float* out = (float*)d_out;

    const int Nn = in_sizes[0] / D_IN;     // 50000
    const int E  = in_sizes[1] / 2;        // 800000
    const int* srcp = ei;
    const int* dstp = ei + E;

    // Workspace layout (all written before read, every call)
    float* H    = (float*)d_ws;                 // Nn x 256 (per-layer XW)
    float* A    = H   + (size_t)Nn * D_H;       // Nn x 256 (aggregated output, reused)
    float* deg  = A   + (size_t)Nn * D_H;       // Nn
    float* dinv = deg + Nn;                     // Nn

    const int TB = 256;

    // Degrees & normalization (shared by all layers)
    gcn_deg_init <<<(Nn + TB - 1) / TB, TB, 0, stream>>>(deg, Nn);
    gcn_deg_count<<<(E  + TB - 1) / TB, TB, 0, stream>>>(dstp, deg, E);
    gcn_dinv     <<<(Nn + TB - 1) / TB, TB, 0, stream>>>(deg, dinv, Nn);

    long nfH = (long)Nn * D_H;
    long nfO = (long)Nn * D_OUT;
    int edgeThreads = E * 32;

    auto gemmBlocks = [&](int Nc) {
        int waves = ((Nn + 15) / 16) * (Nc / 64);
        return (waves * 32 + 127) / 128;
    };

    // ---- Layer 1: h1 = agg(x @ W1) + b1 ----
    gcn_gemm_f32<256, 256, false><<<gemmBlocks(D_H), 128, 0, stream>>>(x, W1, H, Nn);
    gcn_init_out<256><<<(nfH + TB - 1) / TB, TB, 0, stream>>>(H, b1, dinv, A, Nn);
    gcn_scatter<256><<<(edgeThreads + TB - 1) / TB, TB, 0, stream>>>(H, srcp, dstp, dinv, A, E);

    // ---- Layer 2: h2 = agg(relu(h1) @ W2) + b2 ----
    gcn_gemm_f32<256, 256, true><<<gemmBlocks(D_H), 128, 0, stream>>>(A, W2, H, Nn);
    gcn_init_out<256><<<(nfH + TB - 1) / TB, TB, 0, stream>>>(H, b2, dinv, A, Nn);
    gcn_scatter<256><<<(edgeThreads + TB - 1) / TB, TB, 0, stream>>>(H, srcp, dstp, dinv, A, E);

    // ---- Layer 3: out = agg(relu(h2) @ W3) + b3 ----
    gcn_gemm_f32<256, 128, true><<<gemmBlocks(D_OUT), 128, 0, stream>>>(A, W3, H, Nn);
    gcn_init_out<128><<<(nfO + TB - 1) / TB, TB, 0, stream>>>(H, b3, dinv, out, Nn);
    gcn_scatter<128><<<(edgeThreads + TB - 1) / TB, TB, 0, stream>>>(H, srcp, dstp, dinv, out, E);
}